// TransportDelayModule_16269336117703
// MI455X (gfx1250) — compile-verified
//
#include <hip/hip_runtime.h>
#include <hip/hip_bf16.h>

typedef float v2f __attribute__((ext_vector_type(2)));
typedef float v8f __attribute__((ext_vector_type(8)));

#define B_DIM 32
#define T_DIM 168
#define N_DIM 64
#define F_DIM 128
#define TMIN  143          // T-1 - MAX_DELAY_HOURS = 167 - 24
#define XPAD  20           // padded LDS row stride (floats) for bank-conflict avoidance

__global__ __launch_bounds__(128)
void transport_delay_wmma(const float* __restrict__ x_raw,
                          const float* __restrict__ adj,
                          const float* __restrict__ dist,
                          float* __restrict__ out)
{
    const int b    = blockIdx.x >> 3;   // 8 f-tiles per batch
    const int ft   = blockIdx.x & 7;
    const int tid  = threadIdx.x;
    const int wave = tid >> 5;          // 0..3 -> i-tile
    const int lane = tid & 31;
    const int half = lane >> 4;         // lane half selects K offset {0,2}
    const int l16  = lane & 15;

    __shared__ float sSpeed[N_DIM];
    __shared__ float sX[2][N_DIM * XPAD];

    // ---- 1) wind speed per source node j (threads 0..63) ----
    if (tid < N_DIM) {
        const int j = tid;
        float s = 0.0f;
        #pragma unroll
        for (int k = 0; k < 4; ++k)
            s += x_raw[(((size_t)b * T_DIM + (T_DIM - 4 + k)) * N_DIM + j) * F_DIM + 10];
        const float wspm = fmaxf(s * 0.25f * 1.5f + 2.0f, 0.0f);
        sSpeed[j] = wspm * 3.6f + 0.001f;
    }
    __syncthreads();

    // ---- 2) register-cache interpolation coefficients for this lane's (i,j) set ----
    // lane covers i = wave*16 + l16, j = 4k + 2*half + e  for k=0..15, e=0..1
    // t1-clamp (t0==T-1 -> t1==t0) is folded into c0 so the hot loop only needs t0.
    int   aT0[32];
    float aC0[32];
    float aC1[32];
    const int i    = wave * 16 + l16;
    const int joff = half * 2;
    #pragma unroll
    for (int k = 0; k < 16; ++k) {
        #pragma unroll
        for (int e = 0; e < 2; ++e) {
            const int j = 4 * k + joff + e;
            float tau = dist[i * N_DIM + j] / sSpeed[j];
            tau = fminf(fmaxf(tau, 0.0f), 24.0f);
            const float tq = (float)(T_DIM - 1) - tau;
            int t0 = (int)floorf(tq);
            t0 = min(max(t0, 0), T_DIM - 1);
            const float w1 = tq - (float)t0;
            const float a  = adj[((size_t)b * N_DIM + i) * N_DIM + j];
            float c0 = a * (1.0f - w1);
            float c1 = a * w1;
            if (t0 == T_DIM - 1) { c0 += c1; c1 = 0.0f; }  // merged clamp
            aT0[2 * k + e] = t0;
            aC0[2 * k + e] = c0;
            aC1[2 * k + e] = c1;
        }
    }

    // ---- 3) staging addresses: 256 float4 slots of the 64x16 slice, 2 per thread ----
    const float* xb = x_raw + (size_t)b * T_DIM * N_DIM * F_DIM;
    const size_t tstride = (size_t)N_DIM * F_DIM;
    const int idx0 = tid,        r0 = idx0 >> 2, q0 = (idx0 & 3) * 4;
    const int idx1 = tid + 128,  r1 = idx1 >> 2, q1 = (idx1 & 3) * 4;
    const float* g0 = xb + (size_t)r0 * F_DIM + ft * 16 + q0;
    const float* g1 = xb + (size_t)r1 * F_DIM + ft * 16 + q1;

    // prologue: stage slice t = TMIN into buffer 0
    {
        const float4 v0 = *(const float4*)(g0 + (size_t)TMIN * tstride);
        const float4 v1 = *(const float4*)(g1 + (size_t)TMIN * tstride);
        *(float4*)&sX[0][r0 * XPAD + q0] = v0;
        *(float4*)&sX[0][r1 * XPAD + q1] = v1;
    }
    __syncthreads();

    v8f acc = {};

    // ---- 4) out[b] += W_t (64x64) @ x[b,t] (64x16 slice), software-pipelined ----
    for (int t = TMIN; t < T_DIM; ++t) {
        const int buf  = (t - TMIN) & 1;
        const bool more = (t + 1 < T_DIM);

        // issue next slice's global loads before compute (latency hidden by WMMAs)
        float4 n0, n1;
        if (more) {
            n0 = *(const float4*)(g0 + (size_t)(t + 1) * tstride);
            n1 = *(const float4*)(g1 + (size_t)(t + 1) * tstride);
            if (t + 2 < T_DIM) {
                __builtin_prefetch(g0 + (size_t)(t + 2) * tstride, 0, 1);
                __builtin_prefetch(g1 + (size_t)(t + 2) * tstride, 0, 1);
            }
        }

        const int tm1 = t - 1;
        #pragma unroll
        for (int k = 0; k < 16; ++k) {
            v2f afrag, bfrag;
            #pragma unroll
            for (int e = 0; e < 2; ++e) {
                const int t0 = aT0[2 * k + e];
                const float w = ((t   == t0) ? aC0[2 * k + e] : 0.0f)
                              + ((tm1 == t0) ? aC1[2 * k + e] : 0.0f);
                afrag[e] = w;
                bfrag[e] = sX[buf][(4 * k + joff + e) * XPAD + l16];
            }
            acc = __builtin_amdgcn_wmma_f32_16x16x4_f32(false, afrag, false, bfrag,
                                                        (short)0, acc, false, false);
        }

        if (more) {
            *(float4*)&sX[buf ^ 1][r0 * XPAD + q0] = n0;
            *(float4*)&sX[buf ^ 1][r1 * XPAD + q1] = n1;
        }
        __syncthreads();
    }

    // ---- 5) write 16x16 C tile: VGPR r -> row (r + 8*half), lane l16 -> col ----
    #pragma unroll
    for (int r = 0; r < 8; ++r) {
        const int oi = wave * 16 + r + half * 8;
        const int of = ft * 16 + l16;
        out[((size_t)b * N_DIM + oi) * F_DIM + of] = acc[r];
    }
}

extern "C" void kernel_launch(void* const* d_in, const int* in_sizes, int n_in,
                              void* d_out, int out_size, void* d_ws, size_t ws_size,
                              hipStream_t stream) {
    const float* x_raw = (const float*)d_in[0];   // (32,168,64,128) f32
    const float* adj   = (const float*)d_in[1];   // (32,64,64) f32
    const float* dist  = (const float*)d_in[2];   // (64,64) f32
    float* out = (float*)d_out;                   // (32,64,128) f32

    dim3 grid(B_DIM * (F_DIM / 16));              // 256 blocks
    dim3 block(128);                              // 4 wave32s
    transport_delay_wmma<<<grid, block, 0, stream>>>(x_raw, adj, dist, out);
}